// End2EndTongueROI_Dynamic_NMS_50689204027675
// MI455X (gfx1250) — compile-verified
//
#include <hip/hip_runtime.h>
#include <math.h>

typedef __attribute__((ext_vector_type(2))) float v2f;
typedef __attribute__((ext_vector_type(8))) float v8f;

#define N_ANCH   8400
#define NCOLS    37
#define NM       32
#define HMWM     25600      // 160*160
#define IMG      640
#define H0       2160
#define W0       3840
#define NPIX     (H0*W0)    // 8294400
#define MASK_THR 0.72f

// workspace layout in floats
#define OFF_BOX      0
#define OFF_COEF     8
#define OFF_MASKNESS 64
#define OFF_MLOGIT   16384
#define OFF_M640     65536

// ---------------------------------------------------------------------------
// Kernel 1: per-anchor scoring, reductions (max maskness, max box), argmax.
// Single block; 8400 anchors is tiny. Writes final box (xyxy clipped) + coef.
// ---------------------------------------------------------------------------
__global__ __launch_bounds__(256) void k_select(const float* __restrict__ pred,
                                                float* __restrict__ ws) {
    __shared__ float s_a[256];
    __shared__ float s_b[256];
    __shared__ int   s_i[256];
    __shared__ float sh_maskmax, sh_boxmax;
    const int tid = threadIdx.x;

    // pass 1: maskness per anchor + max maskness + max box coord
    float mmax = 0.0f, bmax = -1e30f;
    for (int i = tid; i < N_ANCH; i += 256) {
        const float* row = pred + i * NCOLS;
        float acc = 0.0f;
#pragma unroll
        for (int k = 0; k < NM; ++k) acc += fabsf(row[5 + k]);
        float mn = acc * (1.0f / 32.0f);
        ws[OFF_MASKNESS + i] = mn;
        mmax = fmaxf(mmax, mn);
        bmax = fmaxf(fmaxf(fmaxf(row[0], row[1]), fmaxf(row[2], row[3])), bmax);
    }
    s_a[tid] = mmax; s_b[tid] = bmax;
    __syncthreads();
    for (int s = 128; s > 0; s >>= 1) {
        if (tid < s) {
            s_a[tid] = fmaxf(s_a[tid], s_a[tid + s]);
            s_b[tid] = fmaxf(s_b[tid], s_b[tid + s]);
        }
        __syncthreads();
    }
    if (tid == 0) { sh_maskmax = s_a[0]; sh_boxmax = s_b[0]; }
    __syncthreads();
    const float maskmax = sh_maskmax;
    const bool  is_norm = (sh_boxmax <= 1.2f);

    // pass 2: score + argmax (ties -> lowest index, matching lax.top_k)
    float best = -1e30f; int besti = 0;
    for (int i = tid; i < N_ANCH; i += 256) {
        const float* row = pred + i * NCOLS;
        float sc  = 1.0f / (1.0f + __expf(-row[4]));      // sigmoid(cls), nc==1
        float score = fmaxf(sc - 0.5f, 0.0f) + 0.001f;
        score *= ws[OFF_MASKNESS + i] / (maskmax + 1e-6f);
        float cx = is_norm ? row[0] * 640.0f : row[0];
        float cy = is_norm ? row[1] * 640.0f : row[1];
        float dx = fabsf(cx - 320.0f) * (1.0f / 320.0f);
        float dy = fabsf(cy - 320.0f) * (1.0f / 320.0f);
        float cw = fminf(fmaxf(1.0f - 0.5f * (dx + dy), 0.0f), 1.0f);
        score *= 0.5f + 0.5f * cw;
        if (score > best) { best = score; besti = i; }   // ascending i -> first max kept
    }
    s_a[tid] = best; s_i[tid] = besti;
    __syncthreads();
    for (int s = 128; s > 0; s >>= 1) {
        if (tid < s) {
            float v2 = s_a[tid + s]; int i2 = s_i[tid + s];
            if (v2 > s_a[tid] || (v2 == s_a[tid] && i2 < s_i[tid])) {
                s_a[tid] = v2; s_i[tid] = i2;
            }
        }
        __syncthreads();
    }
    if (tid == 0) {
        const float* row = pred + s_i[0] * NCOLS;
        float cx = row[0], cy = row[1], w = row[2], h = row[3];
        ws[OFF_BOX + 0] = fminf(fmaxf(cx - 0.5f * w, 0.0f), 639.0f);
        ws[OFF_BOX + 1] = fminf(fmaxf(cy - 0.5f * h, 0.0f), 639.0f);
        ws[OFF_BOX + 2] = fminf(fmaxf(cx + 0.5f * w, 0.0f), 639.0f);
        ws[OFF_BOX + 3] = fminf(fmaxf(cy + 0.5f * h, 0.0f), 639.0f);
        for (int k = 0; k < NM; ++k) ws[OFF_COEF + k] = row[5 + k];
    }
}

// ---------------------------------------------------------------------------
// Kernel 2: mask logits = coef(1x32) @ proto(32x25600) via V_WMMA_F32_16X16X4_F32.
// A = coef broadcast over 16 rows; 8 chained K=4 WMMAs accumulate K=32.
// One wave per 16-column tile; lanes 0-15 hold row M=0 of the accumulator.
// ---------------------------------------------------------------------------
__global__ __launch_bounds__(32) void k_maskmm(const float* __restrict__ proto,
                                               float* __restrict__ ws) {
    const int lane  = threadIdx.x;            // 0..31, one wave32
    const int col   = blockIdx.x * 16 + (lane & 15);
    const int khalf = (lane >> 4) * 2;        // lanes 0-15 -> K {0,1}; 16-31 -> {2,3}
    const float* coef = ws + OFF_COEF;

    v8f c = {};
#pragma unroll
    for (int kc = 0; kc < 8; ++kc) {
        const int k0 = kc * 4 + khalf;
        v2f a, b;
        a.x = coef[k0];
        a.y = coef[k0 + 1];
        b.x = proto[k0 * HMWM + col];
        b.y = proto[(k0 + 1) * HMWM + col];
        // D = A(16x4) x B(4x16) + C ; emits v_wmma_f32_16x16x4_f32
        c = __builtin_amdgcn_wmma_f32_16x16x4_f32(false, a, false, b,
                                                  (short)0, c, false, false);
    }
    if (lane < 16)
        ws[OFF_MLOGIT + blockIdx.x * 16 + lane] = c[0];   // row M=0, N=lane
}

// ---------------------------------------------------------------------------
// Kernel 3: bilinear 160->640 (half-pixel centers, edge clamp) + sigmoid.
// ---------------------------------------------------------------------------
__global__ __launch_bounds__(256) void k_up640(float* __restrict__ ws) {
    const int j = blockIdx.x * 256 + threadIdx.x;
    if (j >= IMG * IMG) return;
    const int y = j / IMG, x = j - y * IMG;
    const float* ml = ws + OFF_MLOGIT;

    float xf = (x + 0.5f) * 0.25f - 0.5f;
    float yf = (y + 0.5f) * 0.25f - 0.5f;
    int x0 = (int)floorf(xf); float tx = xf - (float)x0;
    int y0 = (int)floorf(yf); float ty = yf - (float)y0;
    int x0c = min(max(x0, 0), 159), x1c = min(max(x0 + 1, 0), 159);
    int y0c = min(max(y0, 0), 159), y1c = min(max(y0 + 1, 0), 159);
    float v00 = ml[y0c * 160 + x0c], v01 = ml[y0c * 160 + x1c];
    float v10 = ml[y1c * 160 + x0c], v11 = ml[y1c * 160 + x1c];
    float v = (1.0f - ty) * ((1.0f - tx) * v00 + tx * v01)
            +          ty * ((1.0f - tx) * v10 + tx * v11);
    ws[OFF_M640 + j] = 1.0f / (1.0f + __expf(-v));
}

// ---------------------------------------------------------------------------
// Kernel 4: streaming stage. 4 px/thread, float4 I/O, mask shared across 3 ch.
// ~200 MB HBM traffic -> roofline-bound (~8.6us at 23.3 TB/s).
// ---------------------------------------------------------------------------
__global__ __launch_bounds__(256) void k_final(const float* __restrict__ x_raw,
                                               const float* __restrict__ ws,
                                               float* __restrict__ out) {
    const int g = blockIdx.x * 256 + threadIdx.x;
    const int p = g * 4;
    if (p >= NPIX) return;
    const int y = p / W0;
    const int x = p - y * W0;         // W0 % 4 == 0 -> 4 px share a row

    const float fb0 = ws[OFF_BOX + 0] * 6.0f;      // sx = 3840/640
    const float fb1 = ws[OFF_BOX + 1] * 3.375f;    // sy = 2160/640
    const float fb2 = ws[OFF_BOX + 2] * 6.0f;
    const float fb3 = ws[OFF_BOX + 3] * 3.375f;
    const float* m = ws + OFF_M640;

    const float ysf = (float)y;
    const bool  yin = (ysf >= fb1) && (ysf < fb3);
    float yf = (y + 0.5f) * (640.0f / 2160.0f) - 0.5f;
    int y0 = (int)floorf(yf); float ty = yf - (float)y0;
    int y0c = min(max(y0, 0), 639), y1c = min(max(y0 + 1, 0), 639);

    float bm[4];
#pragma unroll
    for (int q = 0; q < 4; ++q) {
        const int xx = x + q;
        float xf = (xx + 0.5f) * (640.0f / 3840.0f) - 0.5f;
        int x0 = (int)floorf(xf); float tx = xf - (float)x0;
        int x0c = min(max(x0, 0), 639), x1c = min(max(x0 + 1, 0), 639);
        float v00 = m[y0c * 640 + x0c], v01 = m[y0c * 640 + x1c];
        float v10 = m[y1c * 640 + x0c], v11 = m[y1c * 640 + x1c];
        float v = (1.0f - ty) * ((1.0f - tx) * v00 + tx * v01)
                +          ty * ((1.0f - tx) * v10 + tx * v11);
        const float xsf = (float)xx;
        const bool in = (v > MASK_THR) && yin && (xsf >= fb0) && (xsf < fb2);
        bm[q] = in ? 1.0f : 0.0f;
    }
#pragma unroll
    for (int ch = 0; ch < 3; ++ch) {
        const float4 vi = *(const float4*)(x_raw + (size_t)ch * NPIX + p);
        float4 vo;
        vo.x = fminf(fmaxf(vi.x * 255.0f, 0.0f), 255.0f) * bm[0];
        vo.y = fminf(fmaxf(vi.y * 255.0f, 0.0f), 255.0f) * bm[1];
        vo.z = fminf(fmaxf(vi.z * 255.0f, 0.0f), 255.0f) * bm[2];
        vo.w = fminf(fmaxf(vi.w * 255.0f, 0.0f), 255.0f) * bm[3];
        *(float4*)(out + (size_t)ch * NPIX + p) = vo;
    }
}

// ---------------------------------------------------------------------------
extern "C" void kernel_launch(void* const* d_in, const int* in_sizes, int n_in,
                              void* d_out, int out_size, void* d_ws, size_t ws_size,
                              hipStream_t stream) {
    const float* x_raw = (const float*)d_in[0];   // (1,3,2160,3840)
    const float* pred  = (const float*)d_in[1];   // (1,8400,37)
    const float* proto = (const float*)d_in[2];   // (1,32,160,160)
    float* out = (float*)d_out;                   // (1,3,2160,3840)
    float* ws  = (float*)d_ws;

    k_select<<<1, 256, 0, stream>>>(pred, ws);
    k_maskmm<<<HMWM / 16, 32, 0, stream>>>(proto, ws);          // 1600 waves
    k_up640<<<(IMG * IMG + 255) / 256, 256, 0, stream>>>(ws);   // 1600 blocks
    k_final<<<(NPIX / 4 + 255) / 256, 256, 0, stream>>>(x_raw, ws, out); // 8100 blocks
}